// HR_SEMobileCapsNet_32315333935552
// MI455X (gfx1250) — compile-verified
//
#include <hip/hip_runtime.h>
#include <math.h>

// 1/sqrt(1+1e-5)
#define BN_INV 0.9999950000374997f

typedef float v2f __attribute__((ext_vector_type(2)));
typedef float v8f __attribute__((ext_vector_type(8)));

static __device__ inline v8f wmma_f32_4(v2f a, v2f b, v8f c) {
  // D = A(16x4,f32) * B(4x16,f32) + C(16x16,f32)  -> v_wmma_f32_16x16x4_f32
  return __builtin_amdgcn_wmma_f32_16x16x4_f32(
      /*neg_a=*/false, a, /*neg_b=*/false, b,
      /*c_mod=*/(short)0, c, /*reuse_a=*/false, /*reuse_b=*/false);
}

static __device__ inline v8f v8f_zero() {
  v8f z;
#pragma unroll
  for (int j = 0; j < 8; ++j) z[j] = 0.0f;
  return z;
}

// ---------------------------------------------------------------------------
// conv1: 1->16 ch, 3x3 s1 p1, + BN + ReLU. Cin=1 => direct VALU conv
// (memory bound; 9 MACs/output). idx = ((b*16+c)*128+y)*128+x
// ---------------------------------------------------------------------------
__global__ __launch_bounds__(256) void conv1_kernel(
    const float* __restrict__ x, const float* __restrict__ w,
    const float* __restrict__ g, const float* __restrict__ beta,
    float* __restrict__ out) {
  int idx = blockIdx.x * 256 + threadIdx.x;
  int xx = idx & 127;
  int yy = (idx >> 7) & 127;
  int c  = (idx >> 14) & 15;
  int b  = idx >> 18;
  const float* xb = x + (size_t)b * 16384;
  float acc = 0.0f;
#pragma unroll
  for (int ty = 0; ty < 3; ++ty) {
    int iy = yy + ty - 1;
    if (iy < 0 || iy > 127) continue;
#pragma unroll
    for (int tx = 0; tx < 3; ++tx) {
      int ix = xx + tx - 1;
      if (ix < 0 || ix > 127) continue;
      acc += w[c * 9 + ty * 3 + tx] * xb[iy * 128 + ix];
    }
  }
  float val = acc * (g[c] * BN_INV) + beta[c];
  out[idx] = fmaxf(val, 0.0f);
}

// ---------------------------------------------------------------------------
// Per-(b,c) spatial mean for SE squeeze. One block per (b,c).
// ---------------------------------------------------------------------------
__global__ __launch_bounds__(256) void mean_hw_kernel(
    const float* __restrict__ in, float* __restrict__ meanout, int HW) {
  int bc = blockIdx.x;
  int tid = threadIdx.x;
  const float* p = in + (size_t)bc * HW;
  float s = 0.0f;
  for (int j = tid; j < HW; j += 256) s += p[j];
#pragma unroll
  for (int off = 16; off; off >>= 1) s += __shfl_xor(s, off, 32);
  __shared__ float red[8];
  if ((tid & 31) == 0) red[tid >> 5] = s;
  __syncthreads();
  if (tid == 0) {
    float t = 0.0f;
#pragma unroll
    for (int wv = 0; wv < 8; ++wv) t += red[wv];
    meanout[bc] = t / (float)HW;
  }
}

// ---------------------------------------------------------------------------
// SE excitation MLP: gate[b,c] = sigmoid(w2 @ relu(w1 @ mean + b1) + b2)
// One thread per batch (tiny: C<=64, HID<=4).
// ---------------------------------------------------------------------------
__global__ void se_gates_kernel(const float* __restrict__ mean,
                                const float* __restrict__ w1,
                                const float* __restrict__ b1,
                                const float* __restrict__ w2,
                                const float* __restrict__ b2,
                                float* __restrict__ gate, int C, int HID) {
  int t = threadIdx.x;  // batch
  float y[4];
  for (int j = 0; j < HID; ++j) {
    float a = b1[j];
    for (int c = 0; c < C; ++c) a += w1[j * C + c] * mean[t * C + c];
    y[j] = fmaxf(a, 0.0f);
  }
  for (int c = 0; c < C; ++c) {
    float a = b2[c];
    for (int j = 0; j < HID; ++j) a += w2[c * HID + j] * y[j];
    gate[t * C + c] = 1.0f / (1.0f + expf(-a));
  }
}

// ---------------------------------------------------------------------------
// Implicit-GEMM 3x3 conv with f32 WMMA (16x16x4).
// One wave computes a [COUT x 16-output-pixels] tile for fixed (b, oy).
//   GEMM: M = COUT (COUT/16 M-tiles), N = 16 pixels, K = CIN*9 (step 4).
// A-fragments = weight rows (aligned float2 from global, L2/WGP$ hot).
// B-fragments = im2col from an LDS-staged input patch; the SE gate of the
// *previous* block and zero-padding are applied during the LDS fill.
// Epilogue: BN scale/bias (+ReLU) or plain bias.
// ---------------------------------------------------------------------------
template <int CIN, int COUT, int STRIDE, bool RELU>
__global__ __launch_bounds__(32) void conv3x3_wmma_kernel(
    const float* __restrict__ in, const float* __restrict__ gate,
    const float* __restrict__ w, const float* __restrict__ bn_g,
    const float* __restrict__ beta, float* __restrict__ out,
    int HIN, int WIN) {
  constexpr int K = CIN * 9;
  constexpr int WPATCH = 15 * STRIDE + 3;
  constexpr int MT = COUT / 16;
  const int HOUT = HIN / STRIDE;
  const int WOUT = WIN / STRIDE;
  const int tilesx = WOUT >> 4;

  int seg = blockIdx.x;
  int txt = seg % tilesx; seg /= tilesx;
  int oy = seg % HOUT;
  int b  = seg / HOUT;
  int ox0 = txt << 4;
  int lane = threadIdx.x;

  __shared__ float patch[CIN * 3 * WPATCH];
  const int iy0 = oy * STRIDE - 1;
  const int ix0 = ox0 * STRIDE - 1;
  for (int idx = lane; idx < CIN * 3 * WPATCH; idx += 32) {
    int cc = idx % WPATCH;
    int t  = idx / WPATCH;
    int r = t % 3, cin = t / 3;
    int iy = iy0 + r, ix = ix0 + cc;
    float vv = 0.0f;
    if (iy >= 0 && iy < HIN && ix >= 0 && ix < WIN) {
      vv = in[((size_t)(b * CIN + cin) * HIN + iy) * WIN + ix];
      vv *= gate[b * CIN + cin];
    }
    patch[idx] = vv;
  }
  __syncthreads();

  v8f acc[MT];
#pragma unroll
  for (int m = 0; m < MT; ++m) acc[m] = v8f_zero();

  const int laneHalf = lane >> 4;   // 0: K={k,k+1}  1: K={k+2,k+3}
  const int n = lane & 15;          // output pixel / weight row within tile

  for (int k = 0; k < K; k += 4) {
    int kk = k + 2 * laneHalf;
    int cin0 = kk / 9;       int tap0 = kk - cin0 * 9;
    int cin1 = (kk + 1) / 9; int tap1 = (kk + 1) - cin1 * 9;
    v2f bfrag;
    bfrag.x = patch[(cin0 * 3 + tap0 / 3) * WPATCH + n * STRIDE + tap0 % 3];
    bfrag.y = patch[(cin1 * 3 + tap1 / 3) * WPATCH + n * STRIDE + tap1 % 3];
#pragma unroll
    for (int m = 0; m < MT; ++m) {
      const float* wp = w + (size_t)(m * 16 + n) * K + kk;  // 8B aligned
      v2f afrag = *reinterpret_cast<const v2f*>(wp);
      acc[m] = wmma_f32_4(afrag, bfrag, acc[m]);
    }
  }

  // D layout: lane L, vgpr v -> row = v + 8*(L>=16), col = L&15
#pragma unroll
  for (int m = 0; m < MT; ++m) {
#pragma unroll
    for (int v = 0; v < 8; ++v) {
      int cout = m * 16 + 8 * laneHalf + v;
      float sc = bn_g ? (bn_g[cout] * BN_INV) : 1.0f;
      float val = acc[m][v] * sc + beta[cout];
      if (RELU) val = fmaxf(val, 0.0f);
      out[((size_t)(b * COUT + cout) * HOUT + oy) * WOUT + ox0 + n] = val;
    }
  }
}

// ---------------------------------------------------------------------------
// Primary caps: reshape (B,64,16,16) -> capsules (B,2048,8) + squash.
// i = (c/8)*256 + h*16 + w ; e = c%8
// ---------------------------------------------------------------------------
__global__ __launch_bounds__(256) void squash_u_kernel(
    const float* __restrict__ ur, float* __restrict__ u) {
  int idx = blockIdx.x * 256 + threadIdx.x;  // b*2048 + i
  int b = idx >> 11, i = idx & 2047;
  int a = i >> 8, r = i & 255;
  const float* base = ur + ((size_t)(b * 64 + a * 8)) * 256 + r;
  float s[8];
  float n2 = 0.0f;
#pragma unroll
  for (int e = 0; e < 8; ++e) {
    s[e] = base[(size_t)e * 256];
    n2 += s[e] * s[e];
  }
  float nn = sqrtf(n2);
  float f = (n2 / (1.0f + n2)) / (nn + 1e-8f);
  float* o = u + (size_t)idx * 8;
#pragma unroll
  for (int e = 0; e < 8; ++e) o[e] = s[e] * f;
}

// ---------------------------------------------------------------------------
// u_hat[b,i,c,d] = sum_e W[i,c,d,e] * u[b,i,e]
// Per capsule i: (48x8) @ (8x64 batches). One wave per (i, 16-batch tile):
// 3 M-tiles x 2 K-steps = 6 WMMAs. Block of 128 = 4 batch tiles of capsule i.
// ---------------------------------------------------------------------------
__global__ __launch_bounds__(128) void uhat_wmma_kernel(
    const float* __restrict__ Wd,   // (2048,3,16,8) = 2048 x 384
    const float* __restrict__ u,    // (64,2048,8)
    float* __restrict__ uhat) {     // (64,2048,48)
  int i = blockIdx.x;
  int ntile = threadIdx.x >> 5;
  int lane = threadIdx.x & 31;
  int laneHalf = lane >> 4;
  int n = lane & 15;
  int batch = ntile * 16 + n;

  v8f acc[3];
#pragma unroll
  for (int m = 0; m < 3; ++m) acc[m] = v8f_zero();

  const float* wbase = Wd + (size_t)i * 384;
#pragma unroll
  for (int k = 0; k < 8; k += 4) {
    int kk = k + 2 * laneHalf;
    v2f bfrag = *reinterpret_cast<const v2f*>(
        u + ((size_t)batch * 2048 + i) * 8 + kk);
#pragma unroll
    for (int m = 0; m < 3; ++m) {
      v2f afrag = *reinterpret_cast<const v2f*>(wbase + (m * 16 + n) * 8 + kk);
      acc[m] = wmma_f32_4(afrag, bfrag, acc[m]);
    }
  }

  float* obase = uhat + ((size_t)batch * 2048 + i) * 48;
#pragma unroll
  for (int m = 0; m < 3; ++m) {
#pragma unroll
    for (int v = 0; v < 8; ++v)
      obase[m * 16 + 8 * laneHalf + v] = acc[m][v];  // 8 contiguous floats
  }
}

// ---------------------------------------------------------------------------
// One routing iteration. Block per batch (u_hat slice is 384KB/batch, L2 hot).
// iter 0: c uniform (b==0), no b_io access.
// iter 1: b_new = dot(u_hat, v_in)           (written, since b was 0)
// iter 2: b_new = b_io + dot(u_hat, v_in)    (last iter: no write, emits out)
// ---------------------------------------------------------------------------
__global__ __launch_bounds__(256) void routing_kernel(
    const float* __restrict__ uhat, float* __restrict__ b_io,
    const float* __restrict__ v_in, float* __restrict__ v_out,
    float* __restrict__ out_norm, int iter) {
  int b = blockIdx.x;
  int tid = threadIdx.x;
  int lane = tid & 31, wv = tid >> 5;

  __shared__ float vin_s[48];
  __shared__ float red[8][48];
  __shared__ float s_fin[48];
  if (iter > 0 && tid < 48) vin_s[tid] = v_in[b * 48 + tid];
  __syncthreads();

  float acc[48];
#pragma unroll
  for (int t = 0; t < 48; ++t) acc[t] = 0.0f;

  for (int i = tid; i < 2048; i += 256) {
    const float4* up4 =
        reinterpret_cast<const float4*>(uhat + ((size_t)b * 2048 + i) * 48);
    float uh[48];
#pragma unroll
    for (int q = 0; q < 12; ++q) {
      float4 t4 = up4[q];
      uh[4 * q + 0] = t4.x; uh[4 * q + 1] = t4.y;
      uh[4 * q + 2] = t4.z; uh[4 * q + 3] = t4.w;
    }
    float cf[3];
    if (iter == 0) {
      cf[0] = cf[1] = cf[2] = (1.0f / 3.0f);
    } else {
      float bv[3];
#pragma unroll
      for (int c = 0; c < 3; ++c) {
        float dot = 0.0f;
#pragma unroll
        for (int d = 0; d < 16; ++d) dot += uh[c * 16 + d] * vin_s[c * 16 + d];
        bv[c] = dot + (iter == 2 ? b_io[((size_t)b * 2048 + i) * 3 + c] : 0.0f);
      }
      if (iter == 1) {
#pragma unroll
        for (int c = 0; c < 3; ++c) b_io[((size_t)b * 2048 + i) * 3 + c] = bv[c];
      }
      float mx = fmaxf(bv[0], fmaxf(bv[1], bv[2]));
      float e0 = expf(bv[0] - mx), e1 = expf(bv[1] - mx), e2 = expf(bv[2] - mx);
      float inv = 1.0f / (e0 + e1 + e2);
      cf[0] = e0 * inv; cf[1] = e1 * inv; cf[2] = e2 * inv;
    }
#pragma unroll
    for (int c = 0; c < 3; ++c)
#pragma unroll
      for (int d = 0; d < 16; ++d) acc[c * 16 + d] += cf[c] * uh[c * 16 + d];
  }

#pragma unroll
  for (int t = 0; t < 48; ++t)
#pragma unroll
    for (int off = 16; off; off >>= 1) acc[t] += __shfl_xor(acc[t], off, 32);
  if (lane == 0) {
#pragma unroll
    for (int t = 0; t < 48; ++t) red[wv][t] = acc[t];
  }
  __syncthreads();
  if (tid < 48) {
    float s = 0.0f;
#pragma unroll
    for (int w8 = 0; w8 < 8; ++w8) s += red[w8][tid];
    s_fin[tid] = s;
  }
  __syncthreads();
  if (tid < 3) {
    int c = tid;
    float n2 = 0.0f;
#pragma unroll
    for (int d = 0; d < 16; ++d) { float sv = s_fin[c * 16 + d]; n2 += sv * sv; }
    float nn = sqrtf(n2);
    float f = (n2 / (1.0f + n2)) / (nn + 1e-8f);
#pragma unroll
    for (int d = 0; d < 16; ++d) v_out[b * 48 + c * 16 + d] = s_fin[c * 16 + d] * f;
    if (out_norm) out_norm[b * 3 + c] = nn * f;  // ||squash(s)|| = f*||s||
  }
}

// ---------------------------------------------------------------------------
extern "C" void kernel_launch(void* const* d_in, const int* in_sizes, int n_in,
                              void* d_out, int out_size, void* d_ws,
                              size_t ws_size, hipStream_t stream) {
  (void)in_sizes; (void)n_in; (void)out_size; (void)ws_size;
  const float* x        = (const float*)d_in[0];
  const float* conv1_w  = (const float*)d_in[1];
  const float* bn1_g    = (const float*)d_in[2];
  const float* bn1_b    = (const float*)d_in[3];
  const float* se1_w1   = (const float*)d_in[4];
  const float* se1_b1   = (const float*)d_in[5];
  const float* se1_w2   = (const float*)d_in[6];
  const float* se1_b2   = (const float*)d_in[7];
  const float* conv2_w  = (const float*)d_in[8];
  const float* bn2_g    = (const float*)d_in[9];
  const float* bn2_b    = (const float*)d_in[10];
  const float* se2_w1   = (const float*)d_in[11];
  const float* se2_b1   = (const float*)d_in[12];
  const float* se2_w2   = (const float*)d_in[13];
  const float* se2_b2   = (const float*)d_in[14];
  const float* conv3_w  = (const float*)d_in[15];
  const float* bn3_g    = (const float*)d_in[16];
  const float* bn3_b    = (const float*)d_in[17];
  const float* se3_w1   = (const float*)d_in[18];
  const float* se3_b1   = (const float*)d_in[19];
  const float* se3_w2   = (const float*)d_in[20];
  const float* se3_b2   = (const float*)d_in[21];
  const float* pc_w     = (const float*)d_in[22];
  const float* pc_b     = (const float*)d_in[23];
  const float* W_digit  = (const float*)d_in[24];

  const int B = 64;
  float* p = (float*)d_ws;
  float* h1    = p;                 p += (size_t)16777216;  // (64,16,128,128)
  float* h2    = p;                 p += (size_t)8388608;   // (64,32,64,64)
  float* h3    = p;                 p += (size_t)4194304;   // (64,64,32,32)
  float* u_raw = p;                 p += (size_t)1048576;   // (64,64,16,16)
  float* u     = p;                 p += (size_t)1048576;   // (64,2048,8)
  float* u_hat = p;                 p += (size_t)6291456;   // (64,2048,48)
  float* b_buf = p;                 p += (size_t)393216;    // (64,2048,3)
  float* m_buf = p;                 p += (size_t)4096;
  float* g1    = p;                 p += (size_t)1024;
  float* g2    = p;                 p += (size_t)2048;
  float* g3    = p;                 p += (size_t)4096;
  float* v0    = p;                 p += (size_t)3072;
  float* v1    = p;                 p += (size_t)3072;

  // Block 1 (direct conv) + SE squeeze/excite
  conv1_kernel<<<(B * 16 * 128 * 128) / 256, 256, 0, stream>>>(
      x, conv1_w, bn1_g, bn1_b, h1);
  mean_hw_kernel<<<B * 16, 256, 0, stream>>>(h1, m_buf, 128 * 128);
  se_gates_kernel<<<1, B, 0, stream>>>(m_buf, se1_w1, se1_b1, se1_w2, se1_b2,
                                       g1, 16, 1);

  // Block 2: WMMA implicit GEMM (gate1 folded into input load)
  conv3x3_wmma_kernel<16, 32, 2, true>
      <<<B * 64 * 4, 32, 0, stream>>>(h1, g1, conv2_w, bn2_g, bn2_b, h2, 128, 128);
  mean_hw_kernel<<<B * 32, 256, 0, stream>>>(h2, m_buf, 64 * 64);
  se_gates_kernel<<<1, B, 0, stream>>>(m_buf, se2_w1, se2_b1, se2_w2, se2_b2,
                                       g2, 32, 2);

  // Block 3
  conv3x3_wmma_kernel<32, 64, 2, true>
      <<<B * 32 * 2, 32, 0, stream>>>(h2, g2, conv3_w, bn3_g, bn3_b, h3, 64, 64);
  mean_hw_kernel<<<B * 64, 256, 0, stream>>>(h3, m_buf, 32 * 32);
  se_gates_kernel<<<1, B, 0, stream>>>(m_buf, se3_w1, se3_b1, se3_w2, se3_b2,
                                       g3, 64, 4);

  // Primary caps conv (bias only, no BN/ReLU, gate3 on input)
  conv3x3_wmma_kernel<64, 64, 2, false>
      <<<B * 16 * 1, 32, 0, stream>>>(h3, g3, pc_w, nullptr, pc_b, u_raw, 32, 32);

  // Capsule reshape + squash, then u_hat via WMMA
  squash_u_kernel<<<(B * 2048) / 256, 256, 0, stream>>>(u_raw, u);
  uhat_wmma_kernel<<<2048, 128, 0, stream>>>(W_digit, u, u_hat);

  // Dynamic routing (3 iterations); last writes final class norms.
  routing_kernel<<<B, 256, 0, stream>>>(u_hat, b_buf, nullptr, v0, nullptr, 0);
  routing_kernel<<<B, 256, 0, stream>>>(u_hat, b_buf, v0, v1, nullptr, 1);
  routing_kernel<<<B, 256, 0, stream>>>(u_hat, b_buf, v1, v0, (float*)d_out, 2);
}